// ModulatedDeformConvWithOff_38989713113186
// MI455X (gfx1250) — compile-verified
//
#include <hip/hip_runtime.h>
#include <hip/hip_bf16.h>

typedef __attribute__((ext_vector_type(2))) float v2f;
typedef __attribute__((ext_vector_type(8))) float v8f;

#define BSZ   8
#define CIN   256
#define COUT  256
#define HH    64
#define WW    64
#define HWSZ  (HH * WW)      // 4096
#define KKT   9              // 3x3 taps
#define KTOT  (CIN * KKT)    // 2304

// ---------------------------------------------------------------------------
// Kernel A: offset/mask conv  om[b, 0..26, y, x] = conv3x3(x, w_om) + b_om
// one thread per output pixel; weights staged to LDS in 16-channel chunks
// ---------------------------------------------------------------------------
__global__ __launch_bounds__(256) void om_conv_kernel(
    const float* __restrict__ x, const float* __restrict__ w_om,
    const float* __restrict__ b_om, float* __restrict__ om) {
  __shared__ float sw[16 * 27 * KKT];   // 3888 floats = 15.2 KB

  const int t = threadIdx.x;
  const int p = blockIdx.x * 256 + t;   // 0 .. 32767
  const int b = p >> 12;
  const int pix = p & (HWSZ - 1);
  const int y = pix >> 6;
  const int xx = pix & 63;

  float acc[27];
#pragma unroll
  for (int oc = 0; oc < 27; ++oc) acc[oc] = b_om[oc];

  const float* xb = x + (size_t)b * CIN * HWSZ;

  for (int c0 = 0; c0 < CIN; c0 += 16) {
    __syncthreads();
    for (int i = t; i < 16 * 27 * KKT; i += 256) {
      const int cc = i / (27 * KKT);
      const int r = i % (27 * KKT);          // oc*9 + k
      const int oc = r / KKT;
      const int k = r % KKT;
      sw[i] = w_om[((size_t)oc * CIN + (c0 + cc)) * KKT + k];
    }
    __syncthreads();

    for (int cc = 0; cc < 16; ++cc) {
      const float* xc = xb + (size_t)(c0 + cc) * HWSZ;
      float patch[KKT];
#pragma unroll
      for (int k = 0; k < KKT; ++k) {
        const int ki = k / 3, kj = k % 3;
        const int yy = y - 1 + ki;
        const int xx2 = xx - 1 + kj;
        const bool v = (yy >= 0) && (yy < HH) && (xx2 >= 0) && (xx2 < WW);
        patch[k] = v ? xc[yy * WW + xx2] : 0.0f;
      }
      const float* wc = &sw[cc * 27 * KKT];
#pragma unroll
      for (int oc = 0; oc < 27; ++oc) {
#pragma unroll
        for (int k = 0; k < KKT; ++k)
          acc[oc] = fmaf(patch[k], wc[oc * KKT + k], acc[oc]);
      }
    }
  }

  float* ob = om + (size_t)b * 27 * HWSZ + pix;
#pragma unroll
  for (int oc = 0; oc < 27; ++oc) ob[(size_t)oc * HWSZ] = acc[oc];
}

// ---------------------------------------------------------------------------
// Kernel B: fused deformable sampling + GEMM via V_WMMA_F32_16X16X4_F32
//   out[b, o, p] = sum_{c,kk} cols[b, c, kk, p] * w_dcn[o, c, kk] + b_dcn[o]
// workgroup = 256 threads (8 waves): M=256 x N=32 output tile.
// Double-buffered cols tile: gathers for chunk i+1 issue under the WMMA
// shadow of chunk i (one barrier per chunk).
// ---------------------------------------------------------------------------
#define WPT 32               // pixels (N) per workgroup
#define CC  8                // channels per K-chunk (one per wave)
#define KC  (CC * KKT)       // 72 K-values per chunk
#define NCHUNK (CIN / CC)    // 32 chunks
#define NMETA (WPT * KKT)    // 288 sampling entries

__global__ __launch_bounds__(256) void dcn_wmma_kernel(
    const float* __restrict__ x, const float* __restrict__ om,
    const float* __restrict__ w_dcn, const float* __restrict__ b_dcn,
    float* __restrict__ out) {
  __shared__ float s_w[4][NMETA];          // bilinear weights * mask (0 if OOB)
  __shared__ int s_idx[4][NMETA];          // clamped y*W+x per corner
  __shared__ float s_cols[2][WPT][KC + 2]; // double-buffered B tile, stride 74

  const int tile = blockIdx.x;          // 0 .. 1023
  const int b = tile >> 7;              // 128 tiles per image
  const int p0 = (tile & 127) * WPT;    // pixel offset within image
  const int t = threadIdx.x;
  const int lane = t & 31;
  const int wave = t >> 5;              // 0..7

  // ---- Stage 0: bilinear sampling metadata (per pixel x tap, channel-invariant)
  const float* omb = om + (size_t)b * 27 * HWSZ;
  for (int e = t; e < NMETA; e += 256) {
    const int p = e / KKT;
    const int kk = e % KKT;
    const int pix = p0 + p;
    const int y = pix >> 6;
    const int xx = pix & 63;
    const int ki = kk / 3, kj = kk % 3;

    const float dy = omb[(size_t)kk * HWSZ + pix];
    const float dx = omb[(size_t)(KKT + kk) * HWSZ + pix];
    float mk = omb[(size_t)(2 * KKT + kk) * HWSZ + pix];
    mk = 1.0f / (1.0f + __expf(-mk));   // sigmoid mask, folded into weights

    const float py = (float)(y - 1 + ki) + dy;
    const float px = (float)(xx - 1 + kj) + dx;
    const float fy = floorf(py);
    const float fx = floorf(px);
    const int y0 = (int)fy;
    const int x0 = (int)fx;
    const float ly = py - fy;
    const float lx = px - fx;

    const int y0c = min(max(y0, 0), HH - 1);
    const int y1c = min(max(y0 + 1, 0), HH - 1);
    const int x0c = min(max(x0, 0), WW - 1);
    const int x1c = min(max(x0 + 1, 0), WW - 1);
    const bool vy0 = (y0 >= 0) && (y0 < HH);
    const bool vy1 = (y0 + 1 >= 0) && (y0 + 1 < HH);
    const bool vx0 = (x0 >= 0) && (x0 < WW);
    const bool vx1 = (x0 + 1 >= 0) && (x0 + 1 < WW);

    s_w[0][e] = (vy0 && vx0) ? (1.0f - ly) * (1.0f - lx) * mk : 0.0f;
    s_w[1][e] = (vy0 && vx1) ? (1.0f - ly) * lx * mk : 0.0f;
    s_w[2][e] = (vy1 && vx0) ? ly * (1.0f - lx) * mk : 0.0f;
    s_w[3][e] = (vy1 && vx1) ? ly * lx * mk : 0.0f;
    s_idx[0][e] = y0c * WW + x0c;
    s_idx[1][e] = y0c * WW + x1c;
    s_idx[2][e] = y1c * WW + x0c;
    s_idx[3][e] = y1c * WW + x1c;
  }

  const float* xb = x + (size_t)b * CIN * HWSZ;

  // gather/blend one channel-chunk into cols buffer `buf`
  // wave -> channel (c0 + wave), lane -> pixel
  auto build_cols = [&](int c0, int buf) {
    const float* xc = xb + (size_t)(c0 + wave) * HWSZ;
#pragma unroll
    for (int kk = 0; kk < KKT; ++kk) {
      const int e = lane * KKT + kk;
      const float v = s_w[0][e] * xc[s_idx[0][e]] +
                      s_w[1][e] * xc[s_idx[1][e]] +
                      s_w[2][e] * xc[s_idx[2][e]] +
                      s_w[3][e] * xc[s_idx[3][e]];
      s_cols[buf][lane][wave * KKT + kk] = v;
    }
  };

  __syncthreads();          // metadata visible to all waves
  build_cols(0, 0);
  __syncthreads();          // first cols tile ready

  // ---- accumulators: wave owns M-subtiles {wave, wave+8}, N-subtiles {0,1}
  v8f acc[4] = {};  // [mh*2 + nt]

  const int krow2 = 2 * (lane >> 4);    // K pair selector within a 4-wide step
  const int mrow = lane & 15;           // A/B/C lane row/col index

  // A-operand row bases (fp32 weights, contiguous in (c,kk))
  const float* w_row0 = w_dcn + (size_t)(wave * 16 + mrow) * KTOT;
  const float* w_row1 = w_row0 + (size_t)128 * KTOT;   // +8 M-tiles

  for (int ci = 0; ci < NCHUNK; ++ci) {
    const int c0 = ci * CC;
    const int buf = ci & 1;

    // prefetch-build the next chunk into the other buffer (issues its
    // scattered loads ahead of this chunk's WMMA block)
    if (ci + 1 < NCHUNK) build_cols(c0 + CC, buf ^ 1);

    const float* wa0 = w_row0 + c0 * KKT;   // even float offset -> 8B aligned
    const float* wa1 = w_row1 + c0 * KKT;
#pragma unroll
    for (int s = 0; s < KC / 4; ++s) {      // 18 WMMA K-steps of 4
      const int kb = 4 * s + krow2;         // even
      const v2f a0 = *(const v2f*)(wa0 + kb);
      const v2f a1 = *(const v2f*)(wa1 + kb);
      const v2f b0 = *(const v2f*)(&s_cols[buf][mrow][kb]);        // N tile 0
      const v2f b1 = *(const v2f*)(&s_cols[buf][16 + mrow][kb]);   // N tile 1
      acc[0] = __builtin_amdgcn_wmma_f32_16x16x4_f32(false, a0, false, b0,
                                                     (short)0, acc[0], false, false);
      acc[1] = __builtin_amdgcn_wmma_f32_16x16x4_f32(false, a0, false, b1,
                                                     (short)0, acc[1], false, false);
      acc[2] = __builtin_amdgcn_wmma_f32_16x16x4_f32(false, a1, false, b0,
                                                     (short)0, acc[2], false, false);
      acc[3] = __builtin_amdgcn_wmma_f32_16x16x4_f32(false, a1, false, b1,
                                                     (short)0, acc[3], false, false);
    }
    __syncthreads();   // next buffer fully built / this buffer fully consumed
  }

  // ---- epilogue: D layout lane L, VGPR v -> M = v + 8*(L/16), N = L%16
  float* outb = out + (size_t)b * COUT * HWSZ + p0;
  const int moff = 8 * (lane >> 4);
#pragma unroll
  for (int mh = 0; mh < 2; ++mh) {
#pragma unroll
    for (int nt = 0; nt < 2; ++nt) {
      const int col = nt * 16 + mrow;
#pragma unroll
      for (int v = 0; v < 8; ++v) {
        const int o = (wave + 8 * mh) * 16 + v + moff;
        outb[(size_t)o * HWSZ + col] = acc[mh * 2 + nt][v] + b_dcn[o];
      }
    }
  }
}

// ---------------------------------------------------------------------------
extern "C" void kernel_launch(void* const* d_in, const int* in_sizes, int n_in,
                              void* d_out, int out_size, void* d_ws, size_t ws_size,
                              hipStream_t stream) {
  const float* x     = (const float*)d_in[0];
  const float* w_om  = (const float*)d_in[1];
  const float* b_om  = (const float*)d_in[2];
  const float* w_dcn = (const float*)d_in[3];
  const float* b_dcn = (const float*)d_in[4];
  float* out = (float*)d_out;
  float* om  = (float*)d_ws;   // 8 * 27 * 4096 floats = 3.54 MB scratch

  om_conv_kernel<<<BSZ * HWSZ / 256, 256, 0, stream>>>(x, w_om, b_om, om);
  dcn_wmma_kernel<<<BSZ * HWSZ / WPT, 256, 0, stream>>>(x, om, w_dcn, b_dcn, out);
}